// ConcatenateMeanAbsDiff_18640158064908
// MI455X (gfx1250) — compile-verified
//
#include <hip/hip_runtime.h>
#include <hip/hip_bf16.h>
#include <stdint.h>

#define FEAT_D 128
#define WPB 8   // waves (bonds) per 256-thread block

typedef float v4f __attribute__((ext_vector_type(4)));   // native 128-bit vector

// One wave32 per bond row. Lane l owns float4 [4l, 4l+4).
//   out[b,   0:128] = bond[b,:]            (async global->LDS->global DMA, NT)
//   out[b, 128:256] = 0.5*(a0+a1)          (VGPR path, NT stores)
//   out[b, 256:384] = |a0-a1|              (VGPR path, NT stores)
// Atom gathers use default RT hints: the 128 MB atom table fits in the
// 192 MB L2, so we deliberately keep it cache-resident while streaming
// bond/out around it with NT hints.
__global__ __launch_bounds__(256)
void bond_concat_mean_absdiff(const float* __restrict__ atom,
                              const float* __restrict__ bond,
                              const long long* __restrict__ idx,
                              float* __restrict__ out,
                              int n_bond)
{
    __shared__ v4f sbond[WPB * 32];   // 4 KB staging for the async row copy

    const int lane = threadIdx.x & 31;
    const int wave = threadIdx.x >> 5;
    const int b    = blockIdx.x * WPB + wave;   // wave-uniform
    if (b >= n_bond) return;

    // ---------- Part 1 issue: bond row -> LDS via CDNA5 async DMA ----------
    const unsigned  lds_off = (unsigned)(uintptr_t)&sbond[wave * 32 + lane];
    const int       in_off  = (b * FEAT_D + lane * 4) * (int)sizeof(float);      // < 2^28
    const long long out_row = (long long)b * (3 * FEAT_D);
    const int       out_off = (int)((out_row + lane * 4) * (long long)sizeof(float)); // < 2^30

    asm volatile("global_load_async_to_lds_b128 %0, %1, %2 th:TH_LOAD_NT"
                 :: "v"(lds_off), "v"(in_off), "s"((uint64_t)(uintptr_t)bond)
                 : "memory");

    // ---------- Part 2: gather + mean/|diff| (overlaps with the DMA) ----------
    const long long i0 = idx[2 * b + 0];
    const long long i1 = idx[2 * b + 1];

    const v4f a0 = *((const v4f*)(atom + i0 * FEAT_D) + lane);  // global_load_b128, RT
    const v4f a1 = *((const v4f*)(atom + i1 * FEAT_D) + lane);

    v4f mean = 0.5f * (a0 + a1);
    v4f d    = a0 - a1;
    v4f ad;
    ad.x = fabsf(d.x);
    ad.y = fabsf(d.y);
    ad.z = fabsf(d.z);
    ad.w = fabsf(d.w);

    v4f* orow = (v4f*)(out + out_row) + lane;
    __builtin_nontemporal_store(mean, orow + 32);   // out[b,128:256], th:TH_STORE_NT
    __builtin_nontemporal_store(ad,   orow + 64);   // out[b,256:384], th:TH_STORE_NT

    // ---------- Part 1 completion: LDS -> out[b,0:128] ----------
    asm volatile("s_wait_asynccnt 0" ::: "memory");
    asm volatile("global_store_async_from_lds_b128 %0, %1, %2 th:TH_STORE_NT"
                 :: "v"(out_off), "v"(lds_off), "s"((uint64_t)(uintptr_t)out)
                 : "memory");
    // S_ENDPGM's implicit wait-idle covers the outstanding async store.
}

extern "C" void kernel_launch(void* const* d_in, const int* in_sizes, int n_in,
                              void* d_out, int out_size, void* d_ws, size_t ws_size,
                              hipStream_t stream)
{
    const float*     atom = (const float*)d_in[0];      // (N_ATOM, 128) f32
    const float*     bond = (const float*)d_in[1];      // (N_BOND, 128) f32
    const long long* idx  = (const long long*)d_in[2];  // (N_BOND, 2) i64
    float*           out  = (float*)d_out;              // (N_BOND, 384) f32

    const int n_bond = in_sizes[1] / FEAT_D;
    const int blocks = (n_bond + WPB - 1) / WPB;

    bond_concat_mean_absdiff<<<blocks, 32 * WPB, 0, stream>>>(atom, bond, idx, out, n_bond);
}